// NSELoss_35450660061418
// MI455X (gfx1250) — compile-verified
//
#include <hip/hip_runtime.h>

#define S_STATIONS 1024
#define NSE_EPS 1e-6f

typedef float v4f __attribute__((ext_vector_type(4)));
typedef int   v4i __attribute__((ext_vector_type(4)));

// --- Stage 0: zero the global accumulators (sse[1024], cnt[1024]) ---------
__global__ void nse_zero(float* __restrict__ acc) {
    int i = blockIdx.x * blockDim.x + threadIdx.x;
    if (i < 2 * S_STATIONS) acc[i] = 0.0f;
}

// --- Stage 1: streaming pass, LDS-privatized histogram, atomic flush ------
__global__ __launch_bounds__(512)
void nse_accum(const float* __restrict__ yp,
               const float* __restrict__ yt,
               const int*   __restrict__ st,
               float* __restrict__ sse_g,
               float* __restrict__ cnt_g,
               int n) {
    __shared__ float    s_sse[S_STATIONS];
    __shared__ unsigned s_cnt[S_STATIONS];

    for (int i = threadIdx.x; i < S_STATIONS; i += blockDim.x) {
        s_sse[i] = 0.0f;
        s_cnt[i] = 0u;
    }
    __syncthreads();

    const int stride = gridDim.x * blockDim.x;   // total threads
    const int gid    = blockIdx.x * blockDim.x + threadIdx.x;
    const int n4     = n >> 2;                   // vec4 elements

    const v4f* __restrict__ yp4 = (const v4f*)yp;
    const v4f* __restrict__ yt4 = (const v4f*)yt;
    const v4i* __restrict__ st4 = (const v4i*)st;

    for (int i = gid; i < n4; i += stride) {
        // Prefetch two iterations ahead (speculative; OOB silently dropped)
        __builtin_prefetch(&yp4[i + 2 * stride], 0, 0);
        __builtin_prefetch(&yt4[i + 2 * stride], 0, 0);
        __builtin_prefetch(&st4[i + 2 * stride], 0, 0);

        // Single-pass 192MB stream vs 192MB L2: mark non-temporal so the
        // streams don't evict the hot 8KB accumulator region.
        v4f a = __builtin_nontemporal_load(&yp4[i]);
        v4f b = __builtin_nontemporal_load(&yt4[i]);
        v4i s = __builtin_nontemporal_load(&st4[i]);

        float dx = a.x - b.x;
        float dy = a.y - b.y;
        float dz = a.z - b.z;
        float dw = a.w - b.w;

        atomicAdd(&s_sse[s.x], dx * dx);  atomicAdd(&s_cnt[s.x], 1u);
        atomicAdd(&s_sse[s.y], dy * dy);  atomicAdd(&s_cnt[s.y], 1u);
        atomicAdd(&s_sse[s.z], dz * dz);  atomicAdd(&s_cnt[s.z], 1u);
        atomicAdd(&s_sse[s.w], dw * dw);  atomicAdd(&s_cnt[s.w], 1u);
    }

    // Scalar tail (N = 2^24 in practice, so this is normally empty)
    for (int i = (n4 << 2) + gid; i < n; i += stride) {
        float d = yp[i] - yt[i];
        int   s = st[i];
        atomicAdd(&s_sse[s], d * d);
        atomicAdd(&s_cnt[s], 1u);
    }

    __syncthreads();

    // Hierarchical flush: one pair of global atomics per present station.
    for (int i = threadIdx.x; i < S_STATIONS; i += blockDim.x) {
        unsigned c = s_cnt[i];
        if (c != 0u) {
            atomicAdd(&sse_g[i], s_sse[i]);
            atomicAdd(&cnt_g[i], (float)c);
        }
    }
}

// --- Stage 2: per-station normalization + tree reduction to scalar --------
__global__ __launch_bounds__(1024)
void nse_finalize(const float* __restrict__ sse_g,
                  const float* __restrict__ cnt_g,
                  const float* __restrict__ station_std,
                  float* __restrict__ out) {
    __shared__ float r_val[S_STATIONS];
    __shared__ float r_cnt[S_STATIONS];

    const int t = threadIdx.x;
    float cnt  = cnt_g[t];
    float val  = 0.0f;
    float pres = 0.0f;
    if (cnt > 0.0f) {
        float d = station_std[t] + NSE_EPS;
        // mse = sse / max(cnt,1); cnt >= 1 here so max is a no-op
        val  = (sse_g[t] / cnt) / (d * d);
        pres = 1.0f;
    }
    r_val[t] = val;
    r_cnt[t] = pres;
    __syncthreads();

    for (int off = S_STATIONS / 2; off > 0; off >>= 1) {
        if (t < off) {
            r_val[t] += r_val[t + off];
            r_cnt[t] += r_cnt[t + off];
        }
        __syncthreads();
    }
    if (t == 0) {
        out[0] = r_val[0] / fmaxf(r_cnt[0], 1.0f);
    }
}

extern "C" void kernel_launch(void* const* d_in, const int* in_sizes, int n_in,
                              void* d_out, int out_size, void* d_ws, size_t ws_size,
                              hipStream_t stream) {
    const float* y_pred      = (const float*)d_in[0];
    const float* y_true      = (const float*)d_in[1];
    const int*   stations    = (const int*)d_in[2];
    const float* station_std = (const float*)d_in[3];
    float*       out         = (float*)d_out;

    float* sse_g = (float*)d_ws;             // [1024]
    float* cnt_g = sse_g + S_STATIONS;       // [1024]

    const int n = in_sizes[0];

    // Must re-zero accumulators every call: the harness replays without
    // re-poisoning, and stage 1 accumulates into them.
    nse_zero<<<2, 1024, 0, stream>>>(sse_g);

    // 256 blocks x 512 threads (16 wave32s/block): 131072 threads,
    // 32 float4 iterations each for N = 2^24.
    nse_accum<<<256, 512, 0, stream>>>(y_pred, y_true, stations,
                                       sse_g, cnt_g, n);

    nse_finalize<<<1, 1024, 0, stream>>>(sse_g, cnt_g, station_std, out);
}